// MambaBlock_89515708383664
// MI455X (gfx1250) — compile-verified
//
#include <hip/hip_runtime.h>
#include <math.h>
#include <stdint.h>

typedef __attribute__((ext_vector_type(2))) float v2f;
typedef __attribute__((ext_vector_type(8))) float v8f;
typedef int v4i_vs __attribute__((vector_size(16)));   // matches builtin pointee

// ---------------------------------------------------------------------------
// Mamba block constants (match the reference).
// ---------------------------------------------------------------------------
#define MB_B      2
#define MB_L      2048
#define MB_DM     1024
#define MB_DIN    2048          // EXPAND * D_MODEL
#define MB_N      16            // D_STATE
#define MB_DTRANK 64
#define MB_XDBL   96            // DT_RANK + 2*D_STATE

// ---------------------------------------------------------------------------
// CDNA5 async global->LDS copy support (ASYNCcnt-tracked, no VGPR staging).
// Falls back to register staging + ds_store if the builtin is unavailable.
// Address-space conversion via integer casts: flat shared addresses keep the
// LDS offset in addr[31:0] (ISA 10.2), so truncation yields the AS(3) value.
// ---------------------------------------------------------------------------
#if __has_builtin(__builtin_amdgcn_global_load_async_to_lds_b128)
#define USE_ASYNC_LDS 1
#else
#define USE_ASYNC_LDS 0
#endif

#define AS1 __attribute__((address_space(1)))
#define AS3 __attribute__((address_space(3)))

__device__ __forceinline__ void async_cp_b128(const float* g, float* l) {
#if USE_ASYNC_LDS
  __builtin_amdgcn_global_load_async_to_lds_b128(
      (AS1 v4i_vs*)(uintptr_t)g, (AS3 v4i_vs*)(unsigned)(uintptr_t)l, 0, 0);
#else
  (void)g; (void)l;
#endif
}

__device__ __forceinline__ void wait_async0() {
#if __has_builtin(__builtin_amdgcn_s_wait_asynccnt)
  __builtin_amdgcn_s_wait_asynccnt(0);
#else
  asm volatile("s_wait_asynccnt 0x0" ::: "memory");
#endif
}

// ---------------------------------------------------------------------------
// LDS-tiled fp32 GEMM:  C[M,N] = A[M,K] * W[N,K]^T
// Block = 256 threads = 8 waves, block tile 128x128, K chunked by 32.
// LDS row stride 36 floats (144 B): 16-B aligned rows for b128 traffic and
// conflict-free 16-lane fragment reads (36*r mod 64 distinct, r=0..15).
//
// Async path: double-buffered LDS; per chunk: wait_asynccnt(0) -> barrier ->
// issue 8 async b128 copies for chunk k+1 -> 64 WMMAs on chunk k.
// ---------------------------------------------------------------------------
#define BLK  128
#define KC   32
#define LSTR 36

#if USE_ASYNC_LDS
__device__ __forceinline__ void stage_chunk_async(
    const float* Aload, const float* Wload, int koff, int sA, int sW,
    float (*sA_buf)[LSTR], float (*sW_buf)[LSTR], int lr, int lc)
{
  #pragma unroll
  for (int p = 0; p < 4; ++p) {
    async_cp_b128(Aload + koff + (size_t)(32 * p) * sA, &sA_buf[lr + 32 * p][lc]);
    async_cp_b128(Wload + koff + (size_t)(32 * p) * sW, &sW_buf[lr + 32 * p][lc]);
  }
}
#endif

__global__ __launch_bounds__(256)
void wmma_gemm_tiled(const float* __restrict__ A, const float* __restrict__ W,
                     float* __restrict__ C, int M, int N, int K,
                     int sA, int sW, int sC)
{
  const int tid   = threadIdx.x;
  const int lane  = tid & 31;
  const int wv    = tid >> 5;              // wave 0..7
  const int mBase = blockIdx.y * BLK;
  const int nBase = blockIdx.x * BLK;

  const int ml = lane & 15;                // row/col within a 16x16 tile
  const int kh = (lane >> 4) * 2;          // K-half select (16x16x4 layout)

  // Cooperative staging map: 256 threads cover a 32(row) x 32(col) patch per
  // pass, 4 passes cover 128 rows; one float4 / 16 B per thread per pass.
  const int lr = tid >> 3;                 // 0..31
  const int lc = (tid & 7) * 4;            // 0,4,...,28

  const float* Aload = A + (size_t)(mBase + lr) * sA + lc;
  const float* Wload = W + (size_t)(nBase + lr) * sW + lc;

  v8f acc[8] = {};

#if USE_ASYNC_LDS
  __shared__ float sAt[2][BLK][LSTR];
  __shared__ float sWt[2][BLK][LSTR];

  stage_chunk_async(Aload, Wload, 0, sA, sW, sAt[0], sWt[0], lr, lc);

  const int nChunks = K / KC;
  for (int cc = 0; cc < nChunks; ++cc) {
    const int buf = cc & 1;
    wait_async0();            // this wave's pending chunk is in LDS
    __syncthreads();          // all waves' copies done; prev compute finished

    if (cc + 1 < nChunks) {
      stage_chunk_async(Aload, Wload, (cc + 1) * KC, sA, sW,
                        sAt[buf ^ 1], sWt[buf ^ 1], lr, lc);
      if (cc + 2 < nChunks) {
        __builtin_prefetch(Aload + (cc + 2) * KC, 0, 0);
        __builtin_prefetch(Wload + (cc + 2) * KC, 0, 0);
      }
    }

    #pragma unroll
    for (int k = 0; k < KC; k += 4) {
      v2f a = *(const v2f*)&sAt[buf][wv * 16 + ml][k + kh];
      #pragma unroll
      for (int t = 0; t < 8; ++t) {
        v2f b = *(const v2f*)&sWt[buf][t * 16 + ml][k + kh];
        acc[t] = __builtin_amdgcn_wmma_f32_16x16x4_f32(
            false, a, false, b, (short)0, acc[t], false, false);
      }
    }
  }
#else
  // Fallback: register-staged single-buffer pipeline.
  __shared__ float sAt[BLK][LSTR];
  __shared__ float sWt[BLK][LSTR];

  float4 ra[4], rw[4];
  #pragma unroll
  for (int p = 0; p < 4; ++p) {
    ra[p] = *(const float4*)(Aload + (size_t)(32 * p) * sA);
    rw[p] = *(const float4*)(Wload + (size_t)(32 * p) * sW);
  }
  #pragma unroll
  for (int p = 0; p < 4; ++p) {
    *(float4*)&sAt[lr + 32 * p][lc] = ra[p];
    *(float4*)&sWt[lr + 32 * p][lc] = rw[p];
  }
  __syncthreads();

  for (int kc = 0; kc < K; kc += KC) {
    const bool more = (kc + KC) < K;
    if (more) {
      #pragma unroll
      for (int p = 0; p < 4; ++p) {
        ra[p] = *(const float4*)(Aload + (kc + KC) + (size_t)(32 * p) * sA);
        rw[p] = *(const float4*)(Wload + (kc + KC) + (size_t)(32 * p) * sW);
      }
    }
    #pragma unroll
    for (int k = 0; k < KC; k += 4) {
      v2f a = *(const v2f*)&sAt[wv * 16 + ml][k + kh];
      #pragma unroll
      for (int t = 0; t < 8; ++t) {
        v2f b = *(const v2f*)&sWt[t * 16 + ml][k + kh];
        acc[t] = __builtin_amdgcn_wmma_f32_16x16x4_f32(
            false, a, false, b, (short)0, acc[t], false, false);
      }
    }
    __syncthreads();
    if (more) {
      #pragma unroll
      for (int p = 0; p < 4; ++p) {
        *(float4*)&sAt[lr + 32 * p][lc] = ra[p];
        *(float4*)&sWt[lr + 32 * p][lc] = rw[p];
      }
      __syncthreads();
    }
  }
#endif

  const int rowHalf = (lane >> 4) * 8;
  #pragma unroll
  for (int t = 0; t < 8; ++t) {
    const int n = nBase + t * 16 + ml;
    #pragma unroll
    for (int i = 0; i < 8; ++i)
      C[(size_t)(mBase + wv * 16 + rowHalf + i) * sC + n] = acc[t][i];
  }
}

// ---------------------------------------------------------------------------
// Simple (non-LDS) fp32 WMMA GEMM for the small/skinny GEMMs, with register
// double-buffering of the k+4 fragments so loads overlap the WMMAs.
// EPI: 0 = none, 1 = +bias then softplus (dt path).
// ---------------------------------------------------------------------------
template<int TN, int EPI>
__global__ __launch_bounds__(256)
void wmma_gemm_nt(const float* __restrict__ A, const float* __restrict__ W,
                  const float* __restrict__ bias, float* __restrict__ C,
                  int M, int N, int K, int sA, int sW, int sC)
{
  const int lane    = threadIdx.x & 31;
  const int waveId  = blockIdx.x * 8 + (threadIdx.x >> 5);
  const int nGroups = N / (16 * TN);
  const int mTile   = waveId / nGroups;
  const int nGroup  = waveId % nGroups;
  if (mTile >= M / 16) return;             // wave-uniform

  const int mBase = mTile * 16;
  const int ml    = lane & 15;
  const int kh    = (lane >> 4) * 2;

  v8f acc[TN] = {};

  const float* Arow = A + (size_t)(mBase + ml) * sA + kh;
  const float* Wrow[TN];
  #pragma unroll
  for (int t = 0; t < TN; ++t)
    Wrow[t] = W + (size_t)(nGroup * 16 * TN + t * 16 + ml) * sW + kh;

  v2f a = *(const v2f*)(Arow);
  v2f b[TN];
  #pragma unroll
  for (int t = 0; t < TN; ++t) b[t] = *(const v2f*)(Wrow[t]);

  for (int k = 0; k < K; k += 4) {
    v2f an = a;
    v2f bn[TN];
    #pragma unroll
    for (int t = 0; t < TN; ++t) bn[t] = b[t];
    if (k + 4 < K) {
      an = *(const v2f*)(Arow + k + 4);
      #pragma unroll
      for (int t = 0; t < TN; ++t) bn[t] = *(const v2f*)(Wrow[t] + k + 4);
    }
    #pragma unroll
    for (int t = 0; t < TN; ++t)
      acc[t] = __builtin_amdgcn_wmma_f32_16x16x4_f32(
          false, a, false, b[t], (short)0, acc[t], false, false);
    a = an;
    #pragma unroll
    for (int t = 0; t < TN; ++t) b[t] = bn[t];
  }

  const int rowHalf = (lane >> 4) * 8;
  #pragma unroll
  for (int t = 0; t < TN; ++t) {
    const int n  = nGroup * 16 * TN + t * 16 + ml;
    const float bn2 = (EPI == 1) ? bias[n] : 0.0f;
    #pragma unroll
    for (int i = 0; i < 8; ++i) {
      float v = acc[t][i];
      if (EPI == 1) {
        float xx = v + bn2;
        v = (xx > 20.0f) ? xx : log1pf(expf(xx));   // softplus
      }
      C[(size_t)(mBase + rowHalf + i) * sC + n] = v;
    }
  }
}

// ---------------------------------------------------------------------------
// Causal depthwise conv1d (K=4, left zero-pad) + bias + SiLU.
// ---------------------------------------------------------------------------
__global__ __launch_bounds__(256)
void conv_silu_kernel(const float* __restrict__ xz,
                      const float* __restrict__ conv_w,
                      const float* __restrict__ conv_b,
                      float* __restrict__ x_conv)
{
  const int idx = blockIdx.x * blockDim.x + threadIdx.x;
  const int d = idx % MB_DIN;
  const int l = (idx / MB_DIN) % MB_L;
  const int b = idx / (MB_DIN * MB_L);

  const float* base = xz + (size_t)b * MB_L * (2 * MB_DIN);
  float s = conv_b[d];
  #pragma unroll
  for (int k = 0; k < 4; ++k) {
    const int li = l + k - 3;
    const float xv = (li >= 0) ? base[(size_t)li * (2 * MB_DIN) + d] : 0.0f;
    s += xv * conv_w[d * 4 + k];
  }
  const float sig = 1.0f / (1.0f + __expf(-s));
  x_conv[idx] = s * sig;
}

// ---------------------------------------------------------------------------
// Selective scan: one thread per (b, d), h[16] in VGPRs, sequential over L.
// Fused epilogue: y = (sum C.h + u*D) * silu(z).
// ---------------------------------------------------------------------------
__global__ __launch_bounds__(256)
void scan_kernel(const float* __restrict__ dt,
                 const float* __restrict__ x_dbl,
                 const float* __restrict__ x_conv,
                 const float* __restrict__ xz,
                 const float* __restrict__ A_log,
                 const float* __restrict__ D_param,
                 float* __restrict__ y)
{
  const int d = blockIdx.x * blockDim.x + threadIdx.x;   // 0..MB_DIN-1
  const int b = blockIdx.y;

  float A[MB_N];
  #pragma unroll
  for (int n = 0; n < MB_N; ++n) A[n] = -__expf(A_log[d * MB_N + n]);
  const float Dp = D_param[d];

  float h[MB_N];
  #pragma unroll
  for (int n = 0; n < MB_N; ++n) h[n] = 0.0f;

  const float* dtp = dt     + (size_t)b * MB_L * MB_DIN + d;
  const float* up  = x_conv + (size_t)b * MB_L * MB_DIN + d;
  const float* zp  = xz     + (size_t)b * MB_L * (2 * MB_DIN) + MB_DIN + d;
  const float* bc  = x_dbl  + (size_t)b * MB_L * MB_XDBL + MB_DTRANK;
  float*       yp  = y      + (size_t)b * MB_L * MB_DIN + d;

  for (int t = 0; t < MB_L; ++t) {
    const float dtv = dtp[(size_t)t * MB_DIN];
    const float u   = up [(size_t)t * MB_DIN];
    const float zv  = zp [(size_t)t * (2 * MB_DIN)];

    float Bt[MB_N], Ct[MB_N];
    const float4* Bv = (const float4*)(bc + (size_t)t * MB_XDBL);
    const float4* Cv = (const float4*)(bc + (size_t)t * MB_XDBL + MB_N);
    #pragma unroll
    for (int q = 0; q < 4; ++q) {
      float4 v4 = Bv[q];
      Bt[4*q] = v4.x; Bt[4*q+1] = v4.y; Bt[4*q+2] = v4.z; Bt[4*q+3] = v4.w;
      float4 c4 = Cv[q];
      Ct[4*q] = c4.x; Ct[4*q+1] = c4.y; Ct[4*q+2] = c4.z; Ct[4*q+3] = c4.w;
    }

    const float du = dtv * u;
    float acc = 0.0f;
    #pragma unroll
    for (int n = 0; n < MB_N; ++n) {
      h[n] = __expf(dtv * A[n]) * h[n] + du * Bt[n];
      acc += Ct[n] * h[n];
    }

    const float yv  = acc + u * Dp;
    const float sig = 1.0f / (1.0f + __expf(-zv));
    yp[(size_t)t * MB_DIN] = yv * (zv * sig);
  }
}

// ---------------------------------------------------------------------------
// Host-side orchestration.
// ---------------------------------------------------------------------------
extern "C" void kernel_launch(void* const* d_in, const int* in_sizes, int n_in,
                              void* d_out, int out_size, void* d_ws, size_t ws_size,
                              hipStream_t stream) {
  const float* x          = (const float*)d_in[0];
  const float* in_proj_w  = (const float*)d_in[1];
  const float* conv_w     = (const float*)d_in[2];
  const float* conv_b     = (const float*)d_in[3];
  const float* x_proj_w   = (const float*)d_in[4];
  const float* dt_proj_w  = (const float*)d_in[5];
  const float* dt_proj_b  = (const float*)d_in[6];
  const float* A_log      = (const float*)d_in[7];
  const float* D_param    = (const float*)d_in[8];
  const float* out_proj_w = (const float*)d_in[9];
  float* out = (float*)d_out;

  const int M = MB_B * MB_L;  // 4096 token rows

  // Workspace carve-up (fp32 elements).
  float* ws     = (float*)d_ws;
  float* xz     = ws;                                        // M * 4096
  float* x_conv = xz     + (size_t)M * (2 * MB_DIN);         // M * 2048
  float* x_dbl  = x_conv + (size_t)M * MB_DIN;               // M * 96
  float* dtb    = x_dbl  + (size_t)M * MB_XDBL;              // M * 2048
  float* yb     = dtb    + (size_t)M * MB_DIN;               // M * 2048

  // 1) xz = x @ in_proj_w^T      (4096x1024)·(1024x4096)  -- tiled WMMA
  {
    const int N = 2 * MB_DIN, K = MB_DM;
    dim3 grid(N / BLK, M / BLK);
    wmma_gemm_tiled<<<grid, 256, 0, stream>>>(x, in_proj_w, xz,
                                              M, N, K, K, K, N);
  }

  // 2) causal depthwise conv + SiLU
  {
    const int total = M * MB_DIN;
    conv_silu_kernel<<<total / 256, 256, 0, stream>>>(xz, conv_w, conv_b, x_conv);
  }

  // 3) x_dbl = x_conv @ x_proj_w^T   (4096x2048)·(2048x96)  -- TN=3 strip
  {
    const int N = MB_XDBL, K = MB_DIN;
    const int waves = (M / 16) * (N / 48);
    wmma_gemm_nt<3, 0><<<waves / 8, 256, 0, stream>>>(
        x_conv, x_proj_w, nullptr, x_dbl, M, N, K, K, K, N);
  }

  // 4) dt = softplus(x_dbl[:, :64] @ dt_proj_w^T + b)   (4096x64)·(64x2048)
  {
    const int N = MB_DIN, K = MB_DTRANK;
    const int waves = (M / 16) * (N / 64);
    wmma_gemm_nt<4, 1><<<waves / 8, 256, 0, stream>>>(
        x_dbl, dt_proj_w, dt_proj_b, dtb, M, N, K, MB_XDBL, K, N);
  }

  // 5) selective scan + gating epilogue -> yb
  {
    dim3 grid(MB_DIN / 256, MB_B);
    scan_kernel<<<grid, 256, 0, stream>>>(dtb, x_dbl, x_conv, xz, A_log, D_param, yb);
  }

  // 6) out = yb @ out_proj_w^T   (4096x2048)·(2048x1024)  -- tiled WMMA
  {
    const int N = MB_DM, K = MB_DIN;
    dim3 grid(N / BLK, M / BLK);
    wmma_gemm_tiled<<<grid, 256, 0, stream>>>(yb, out_proj_w, out,
                                              M, N, K, K, K, N);
  }
}